// WL4S_27865747817344
// MI455X (gfx1250) — compile-verified
//
#include <hip/hip_runtime.h>

#define NUM_LAYERS 3
#define HT_SIZE (1 << 18)
#define HT_MASK (HT_SIZE - 1)
#define EMPTY_KEY 0xFFFFFFFFFFFFFFFFull

#define TILE_W 64
#define TILE_STRIDE 66  // even stride => 8B-aligned v2f LDS reads; spreads banks

typedef __attribute__((ext_vector_type(2))) float v2f;
typedef __attribute__((ext_vector_type(8))) float v8f;

#if defined(__has_builtin)
#if __has_builtin(__builtin_amdgcn_global_load_async_to_lds_b32)
#define WL_HAVE_ASYNC_LDS 1
#endif
#if __has_builtin(__builtin_amdgcn_s_wait_asynccnt)
#define WL_HAVE_WAIT_ASYNC 1
#endif
#endif

// The async-to-LDS builtin is declared with int-element pointers:
//   (global int* src, shared int* dst, int offset, int cpol)
typedef __attribute__((address_space(1))) int gint;
typedef __attribute__((address_space(3))) int lint;

__device__ __forceinline__ unsigned long long mix64(unsigned long long z) {
    z += 0x9E3779B97F4A7C15ull;
    z = (z ^ (z >> 30)) * 0xBF58476D1CE4E5B9ull;
    z = (z ^ (z >> 27)) * 0x94D049BB133111EBull;
    return z ^ (z >> 31);
}

// meta layout (ints): [0..2] = nc per layer, [3..5] = output offset per layer, [6] = bsize
__global__ void wl_zero_out(float* out, long long n) {
    long long stride = (long long)gridDim.x * blockDim.x;
    for (long long i = (long long)blockIdx.x * blockDim.x + threadIdx.x; i < n; i += stride)
        out[i] = 0.0f;
}

__global__ void wl_init_meta(int* meta) {
    if (threadIdx.x < 16) meta[threadIdx.x] = 0;
}

__global__ void wl_copy_colors(const int* __restrict__ x, int* __restrict__ colors, int n) {
    int i = blockIdx.x * blockDim.x + threadIdx.x;
    if (i < n) colors[i] = x[i];
}

__global__ void wl_bsize(const int* __restrict__ batch, int n, int* meta) {
    int i = blockIdx.x * blockDim.x + threadIdx.x;
    if (i < n) atomicMax(&meta[6], batch[i] + 1);
}

__global__ void wl_layer_clear(unsigned long long* accS, unsigned long long* accX, int n,
                               unsigned long long* htKey, int* htMin, float* norm2) {
    int stride = gridDim.x * blockDim.x;
    int lim = (n > HT_SIZE) ? n : HT_SIZE;
    for (int i = blockIdx.x * blockDim.x + threadIdx.x; i < lim; i += stride) {
        if (i < n) { accS[i] = 0ull; accX[i] = 0ull; }
        if (i < HT_SIZE) { htKey[i] = EMPTY_KEY; htMin[i] = 0x7FFFFFFF; }
        if (i < 1024) norm2[i] = 0.0f;
    }
}

// Stream edges; commutative multiset hash accumulation per destination node.
__global__ void wl_edge_accum(const int* __restrict__ src, const int* __restrict__ dst,
                              long long E, const int* __restrict__ colors,
                              unsigned long long* accS, unsigned long long* accX) {
    long long stride = (long long)gridDim.x * blockDim.x;
    for (long long e = (long long)blockIdx.x * blockDim.x + threadIdx.x; e < E; e += stride) {
        int s = src[e], d = dst[e];
        long long en = e + stride;
        if (en < E) {
            // gfx1250 global_prefetch_b8 for next iteration's random gather
            __builtin_prefetch(&colors[src[en]], 0, 1);
        }
        unsigned long long c = (unsigned long long)(unsigned)colors[s];
        unsigned long long h = mix64(c * 0x9E3779B97F4A7C15ull + 0xD1B54A32D192ED03ull);
        atomicAdd(&accS[d], h);
        atomicXor(&accX[d], mix64(h ^ 0xA24BAED4963EE407ull));
    }
}

// Signature = hash(own color, multiset hash); insert into open-addressed table,
// track min node index per unique signature (first occurrence).
__global__ void wl_sig_insert(const int* __restrict__ colors,
                              const unsigned long long* __restrict__ accS,
                              const unsigned long long* __restrict__ accX, int n,
                              unsigned long long* htKey, int* htMin, int* nodeSlot) {
    int i = blockIdx.x * blockDim.x + threadIdx.x;
    if (i >= n) return;
    unsigned long long sig =
        mix64(((unsigned long long)(unsigned)colors[i] << 1) ^ 0x2545F4914F6CDD1Dull);
    sig = mix64(sig ^ accS[i]);
    sig ^= mix64(accX[i] + 0x632BE59BD9B4E019ull);
    if (sig == EMPTY_KEY) sig = 0x1234567890ABCDEFull;
    unsigned slot = (unsigned)(sig ^ (sig >> 32)) & HT_MASK;
    for (;;) {
        unsigned long long prev = atomicCAS(&htKey[slot], EMPTY_KEY, sig);
        if (prev == EMPTY_KEY || prev == sig) break;
        slot = (slot + 1) & HT_MASK;
    }
    nodeSlot[i] = (int)slot;
    atomicMin(&htMin[slot], i);
}

__global__ void wl_flag_blocksum(const int* __restrict__ htMin, const int* __restrict__ nodeSlot,
                                 int n, int* blockSums) {
    __shared__ int sm[256];
    int i = blockIdx.x * 256 + threadIdx.x;
    int f = 0;
    if (i < n) f = (htMin[nodeSlot[i]] == i) ? 1 : 0;
    sm[threadIdx.x] = f;
    __syncthreads();
    for (int s = 128; s > 0; s >>= 1) {
        if (threadIdx.x < s) sm[threadIdx.x] += sm[threadIdx.x + s];
        __syncthreads();
    }
    if (threadIdx.x == 0) blockSums[blockIdx.x] = sm[0];
}

// Single-block exclusive scan of block sums (NB <= 1024); also records nc + out offset.
__global__ void wl_scan_blocksums(const int* __restrict__ blockSums, int* blockOffsets,
                                  int NB, int* meta, int layer) {
    __shared__ int sm[1024];
    int t = threadIdx.x;
    int v = (t < NB) ? blockSums[t] : 0;
    sm[t] = v;
    __syncthreads();
    for (int s = 1; s < 1024; s <<= 1) {
        int add = (t >= s) ? sm[t - s] : 0;
        __syncthreads();
        sm[t] += add;
        __syncthreads();
    }
    if (t < NB) blockOffsets[t] = sm[t] - v;  // exclusive
    if (t == 0) {
        meta[layer] = sm[1023];  // total = number of colors this layer
        meta[3 + layer] = (layer == 0) ? 0 : meta[3 + layer - 1] + meta[6] * meta[layer - 1];
    }
}

__global__ void wl_rank_assign(const int* __restrict__ htMin, const int* __restrict__ nodeSlot,
                               int n, const int* __restrict__ blockOffsets, int* htRank) {
    __shared__ int sm[256];
    int i = blockIdx.x * 256 + threadIdx.x;
    int f = 0;
    if (i < n) f = (htMin[nodeSlot[i]] == i) ? 1 : 0;
    sm[threadIdx.x] = f;
    __syncthreads();
    for (int s = 1; s < 256; s <<= 1) {
        int add = (threadIdx.x >= s) ? sm[threadIdx.x - s] : 0;
        __syncthreads();
        sm[threadIdx.x] += add;
        __syncthreads();
    }
    if (i < n && f)
        htRank[nodeSlot[i]] = blockOffsets[blockIdx.x] + sm[threadIdx.x] - 1;
}

__global__ void wl_color_update(const int* __restrict__ htRank, const int* __restrict__ nodeSlot,
                                int n, int* colors) {
    int i = blockIdx.x * blockDim.x + threadIdx.x;
    if (i < n) colors[i] = htRank[nodeSlot[i]];
}

__global__ void wl_hist_scatter(const int* __restrict__ colors, const int* __restrict__ batch,
                                const float* __restrict__ w, int n, const int* __restrict__ meta,
                                int layer, float* out) {
    int i = blockIdx.x * blockDim.x + threadIdx.x;
    if (i >= n) return;
    int nc = meta[layer];
    long long off = (long long)meta[3 + layer];
    atomicAdd(&out[off + (long long)batch[i] * nc + colors[i]], w[i]);
}

// Row sum-of-squares via V_WMMA_F32_16X16X4_F32.
// Stage 16 rows x 64 cols into LDS with coalesced async copies
// (global_load_async_to_lds_b32, ASYNCcnt), then feed the 16x4 F32 A-operand
// from LDS: lane L holds row (L&15), K = (L>>4)*2 .. +1 (2 VGPRs = v2f).
// B = all-ones 4x16 so D[m][*] = sum_k a[m][k]; a = h^2 gives row sum-of-squares.
// gridDim.y waves split the k range; partials combined via atomicAdd.
__global__ __launch_bounds__(32)
void wl_norm_reduce(const float* __restrict__ out, const int* __restrict__ meta, int layer,
                    float* norm2) {
    __shared__ float tile[16][TILE_STRIDE];
    const int nc = meta[layer];
    const long long off = (long long)meta[3 + layer];
    const int bsize = meta[6];
    if (nc <= 0 || bsize <= 0) return;

    const int lane = threadIdx.x;          // 0..31, one wave per block
    const int row0 = blockIdx.x * 16;      // 16 rows per wave
    const int m = lane & 15;
    const int koff = (lane >> 4) * 2;      // lanes 16..31 hold K=2,3 (16x4 F32 A layout)
    const int row = row0 + m;
    const float pred = (row < bsize) ? 1.0f : 0.0f;

    // k-stripe for this blockIdx.y, aligned to TILE_W
    int per = ((nc + (int)gridDim.y - 1) / (int)gridDim.y + (TILE_W - 1)) & ~(TILE_W - 1);
    int kBeg = (int)blockIdx.y * per;
    int kEnd = kBeg + per;
    if (kEnd > nc) kEnd = nc;

    const float* gbase = out + off;
    v8f acc0 = {0.f, 0.f, 0.f, 0.f, 0.f, 0.f, 0.f, 0.f};
    v8f acc1 = {0.f, 0.f, 0.f, 0.f, 0.f, 0.f, 0.f, 0.f};
    const v2f ones = {1.0f, 1.0f};

    for (int kt = kBeg; kt < kEnd; kt += TILE_W) {
        // ---- stage tile: 16 rows, 64 consecutive columns, clamped addresses ----
        int c0 = kt + lane;      if (c0 > kEnd - 1) c0 = kEnd - 1;
        int c1 = kt + 32 + lane; if (c1 > kEnd - 1) c1 = kEnd - 1;
#pragma unroll
        for (int r = 0; r < 16; ++r) {
            int gr = row0 + r; if (gr > bsize - 1) gr = bsize - 1;
            const float* g = gbase + (long long)gr * nc;
#if defined(WL_HAVE_ASYNC_LDS)
            __builtin_amdgcn_global_load_async_to_lds_b32(
                (gint*)(g + c0), (lint*)&tile[r][lane], 0, 0);
            __builtin_amdgcn_global_load_async_to_lds_b32(
                (gint*)(g + c1), (lint*)&tile[r][32 + lane], 0, 0);
#else
            tile[r][lane] = g[c0];
            tile[r][32 + lane] = g[c1];
#endif
        }
#if defined(WL_HAVE_ASYNC_LDS)
#if defined(WL_HAVE_WAIT_ASYNC)
        __builtin_amdgcn_s_wait_asynccnt(0);
#else
        asm volatile("s_wait_asynccnt 0x0" ::: "memory");
#endif
#endif
        // ---- consume tile: branchless, uniform control flow for WMMA ----
        int lim = kEnd - kt;                       // uniform across wave
        if (lim >= TILE_W) {
#pragma unroll
            for (int k4 = 0; k4 < TILE_W; k4 += 8) {
                v2f a = *(const v2f*)&tile[m][k4 + koff];
                v2f b = *(const v2f*)&tile[m][k4 + 4 + koff];
                a.x = a.x * a.x * pred;
                a.y = a.y * a.y * pred;
                b.x = b.x * b.x * pred;
                b.y = b.y * b.y * pred;
                acc0 = __builtin_amdgcn_wmma_f32_16x16x4_f32(false, a, false, ones, (short)0,
                                                             acc0, false, false);
                acc1 = __builtin_amdgcn_wmma_f32_16x16x4_f32(false, b, false, ones, (short)0,
                                                             acc1, false, false);
            }
        } else {
            for (int k4 = 0; k4 < lim; k4 += 4) {
                int c = k4 + koff;
                v2f a = *(const v2f*)&tile[m][c];
                a.x = ((kt + c) < kEnd) ? a.x : 0.0f;      // v_cndmask, no branch
                a.y = ((kt + c + 1) < kEnd) ? a.y : 0.0f;
                a.x = a.x * a.x * pred;
                a.y = a.y * a.y * pred;
                acc0 = __builtin_amdgcn_wmma_f32_16x16x4_f32(false, a, false, ones, (short)0,
                                                             acc0, false, false);
            }
        }
    }
#pragma unroll
    for (int r = 0; r < 8; ++r) acc0[r] += acc1[r];

    // D layout: VGPR r -> rows r (lanes 0-15) and 8+r (lanes 16-31); all columns equal.
    if (lane == 0) {
#pragma unroll
        for (int r = 0; r < 8; ++r) {
            int rr = row0 + r;
            if (rr < bsize) atomicAdd(&norm2[rr], acc0[r]);
        }
    } else if (lane == 16) {
#pragma unroll
        for (int r = 0; r < 8; ++r) {
            int rr = row0 + 8 + r;
            if (rr < bsize) atomicAdd(&norm2[rr], acc0[r]);
        }
    }
}

__global__ void wl_scale(float* out, const int* __restrict__ meta, int layer,
                         const float* __restrict__ norm2) {
    int nc = meta[layer];
    long long off = (long long)meta[3 + layer];
    int bsize = meta[6];
    if (nc <= 0 || bsize <= 0) return;
    long long total = (long long)bsize * nc;
    long long stride = (long long)gridDim.x * blockDim.x;
    for (long long i = (long long)blockIdx.x * blockDim.x + threadIdx.x; i < total; i += stride) {
        int row = (int)(i / nc);
        out[off + i] = out[off + i] / sqrtf(norm2[row]);
    }
}

extern "C" void kernel_launch(void* const* d_in, const int* in_sizes, int n_in,
                              void* d_out, int out_size, void* d_ws, size_t ws_size,
                              hipStream_t stream) {
    const int* x = (const int*)d_in[0];
    const int* ei = (const int*)d_in[1];
    const int* batch = (const int*)d_in[2];
    const float* wgt = (const float*)d_in[3];
    float* out = (float*)d_out;

    const int N = in_sizes[0];
    const long long E = (long long)in_sizes[1] / 2;
    const int* src = ei;
    const int* dst = ei + E;

    // Workspace carve-up (256B aligned slices)
    char* ws = (char*)d_ws;
    size_t off = 0;
    auto take = [&](size_t bytes) -> void* {
        void* p = ws + off;
        off += (bytes + 255) & ~(size_t)255;
        return p;
    };
    unsigned long long* accS = (unsigned long long*)take((size_t)N * 8);
    unsigned long long* accX = (unsigned long long*)take((size_t)N * 8);
    int* colors   = (int*)take((size_t)N * 4);
    int* nodeSlot = (int*)take((size_t)N * 4);
    unsigned long long* htKey = (unsigned long long*)take((size_t)HT_SIZE * 8);
    int* htMin  = (int*)take((size_t)HT_SIZE * 4);
    int* htRank = (int*)take((size_t)HT_SIZE * 4);
    int* blockSums    = (int*)take(1024 * 4);
    int* blockOffsets = (int*)take(1024 * 4);
    int* meta     = (int*)take(64 * 4);
    float* norm2  = (float*)take(1024 * 4);
    (void)ws_size; (void)n_in;

    const int NB = (N + 255) / 256;           // per-node block count (<=1024 assumed)
    const int gStream = 2048;                 // grid-stride streaming grid

    wl_zero_out<<<gStream, 256, 0, stream>>>(out, (long long)out_size);
    wl_init_meta<<<1, 32, 0, stream>>>(meta);
    wl_copy_colors<<<NB, 256, 0, stream>>>(x, colors, N);
    wl_bsize<<<NB, 256, 0, stream>>>(batch, N, meta);

    for (int layer = 0; layer < NUM_LAYERS; ++layer) {
        wl_layer_clear<<<gStream, 256, 0, stream>>>(accS, accX, N, htKey, htMin, norm2);
        wl_edge_accum<<<gStream, 256, 0, stream>>>(src, dst, E, colors, accS, accX);
        wl_sig_insert<<<NB, 256, 0, stream>>>(colors, accS, accX, N, htKey, htMin, nodeSlot);
        wl_flag_blocksum<<<NB, 256, 0, stream>>>(htMin, nodeSlot, N, blockSums);
        wl_scan_blocksums<<<1, 1024, 0, stream>>>(blockSums, blockOffsets, NB, meta, layer);
        wl_rank_assign<<<NB, 256, 0, stream>>>(htMin, nodeSlot, N, blockOffsets, htRank);
        wl_color_update<<<NB, 256, 0, stream>>>(htRank, nodeSlot, N, colors);
        wl_hist_scatter<<<NB, 256, 0, stream>>>(colors, batch, wgt, N, meta, layer, out);
        // L2 norm: 32 row-groups x 64 k-splits = 2048 waves of LDS-staged WMMA reduction
        wl_norm_reduce<<<dim3(32, 64, 1), 32, 0, stream>>>(out, meta, layer, norm2);
        wl_scale<<<gStream, 256, 0, stream>>>(out, meta, layer, norm2);
    }
}